// CAttentionBlock_85504208929453
// MI455X (gfx1250) — compile-verified
//
#include <hip/hip_runtime.h>

typedef __attribute__((ext_vector_type(2))) float v2f;
typedef __attribute__((ext_vector_type(8))) float v8f;

#define WSIZE 2
#define NH 8
#define CDIM 256
#define DH 32
#define NTOK 4

#if defined(__HIP_DEVICE_COMPILE__) && defined(__has_builtin)
#  if __has_builtin(__builtin_amdgcn_global_load_async_to_lds_b128)
#    define HAVE_ASYNC_LDS 1
#  endif
#endif

#if HAVE_ASYNC_LDS
typedef int v4i __attribute__((vector_size(16)));
typedef __attribute__((address_space(1))) v4i* gptr128;   // global b128 pointer
typedef __attribute__((address_space(3))) v4i* lptr128;   // LDS b128 pointer
#endif

// ---- DPP helpers (wave32, pure VALU cross-lane) ----
template <int CTRL>
__device__ __forceinline__ float dppmov(float x) {
    int v = __builtin_amdgcn_update_dpp(0, __builtin_bit_cast(int, x),
                                        CTRL, 0xF, 0xF, true);
    return __builtin_bit_cast(float, v);
}
#define DPP_QUAD_XOR1 0xB1   // quad_perm [1,0,3,2]
#define DPP_QUAD_XOR2 0x4E   // quad_perm [2,3,0,1]
#define DPP_HALF_MIRR 0x141  // row_half_mirror
#define DPP_ROW_MIRR  0x140  // row_mirror

// butterfly softmax over each 4-lane quad (all lanes get result)
__device__ __forceinline__ float quad_softmax(float s) {
    float m = fmaxf(s, dppmov<DPP_QUAD_XOR1>(s));
    m = fmaxf(m, dppmov<DPP_QUAD_XOR2>(m));
    float e = __expf(s - m);
    float d = e + dppmov<DPP_QUAD_XOR1>(e);
    d += dppmov<DPP_QUAD_XOR2>(d);
    return e / d;
}

// sum across each 16-lane row (every lane of the row holds the row sum)
__device__ __forceinline__ float row_sum(float x) {
    x += dppmov<DPP_QUAD_XOR1>(x);
    x += dppmov<DPP_QUAD_XOR2>(x);
    x += dppmov<DPP_HALF_MIRR>(x);
    x += dppmov<DPP_ROW_MIRR>(x);
    return x;
}

// One block = one 2x2 window, 256 threads = 8 waves.
__global__ __launch_bounds__(256)
void swin_cross_attn_fused(const float* __restrict__ rp,
                           const float* __restrict__ gp,
                           const float* __restrict__ bp,
                           const float* __restrict__ ip,
                           const float* __restrict__ lnw0, const float* __restrict__ lnb0,
                           const float* __restrict__ lnw1, const float* __restrict__ lnb1,
                           const float* __restrict__ lnw2, const float* __restrict__ lnb2,
                           const float* __restrict__ lnw3, const float* __restrict__ lnb3,
                           float* __restrict__ out,
                           int Bn, int H, int Wd)
{
    __shared__ __align__(16) float sQ[4][NTOK][CDIM];     // [tensor r,g,b,ir][token][chan]
    __shared__ __align__(16) float sP[4][NH][NTOK][NTOK]; // [pair][head][qi][ki]
    __shared__ float2 red2[8][2][NTOK];                   // [wave][half-row][qi] = (sum, sumsq)

    const int tid  = threadIdx.x;
    const int lane = tid & 31;
    const int wave = tid >> 5;
    const int nWw = Wd / WSIZE;
    const int nWh = H / WSIZE;
    const int wid = blockIdx.x;
    const int bi  = wid / (nWh * nWw);
    const int rem = wid % (nWh * nWw);
    const int wh  = rem / nWw;
    const int ww  = rem % nWw;

    int tokBase[NTOK];
#pragma unroll
    for (int t = 0; t < NTOK; ++t) {
        const int hh = wh * WSIZE + (t >> 1);
        const int wp = ww * WSIZE + (t & 1);
        tokBase[t] = (bi * H + hh) * Wd + wp;
    }

    // ---------------- Phase 1: global -> LDS ----------------
    {
        const int tok = tid >> 6;                // 0..3
        const int c4  = (tid & 63) << 2;         // 0..252
        const long base = (long)tokBase[tok] * CDIM + c4;
#if HAVE_ASYNC_LDS
        __builtin_amdgcn_global_load_async_to_lds_b128(
            (gptr128)(rp + base), (lptr128)&sQ[0][tok][c4], 0, 0);
        __builtin_amdgcn_global_load_async_to_lds_b128(
            (gptr128)(gp + base), (lptr128)&sQ[1][tok][c4], 0, 0);
        __builtin_amdgcn_global_load_async_to_lds_b128(
            (gptr128)(bp + base), (lptr128)&sQ[2][tok][c4], 0, 0);
        __builtin_amdgcn_global_load_async_to_lds_b128(
            (gptr128)(ip + base), (lptr128)&sQ[3][tok][c4], 0, 0);
#  if __has_builtin(__builtin_amdgcn_s_wait_asynccnt)
        __builtin_amdgcn_s_wait_asynccnt(0);
#  else
        asm volatile("s_wait_asynccnt 0x0" ::: "memory");
#  endif
#else
        const float4 vr = *(const float4*)(rp + base);
        const float4 vg = *(const float4*)(gp + base);
        const float4 vb = *(const float4*)(bp + base);
        const float4 vi = *(const float4*)(ip + base);
        *(float4*)&sQ[0][tok][c4] = vr;
        *(float4*)&sQ[1][tok][c4] = vg;
        *(float4*)&sQ[2][tok][c4] = vb;
        *(float4*)&sQ[3][tok][c4] = vi;
#endif
    }
    __syncthreads();

    // ---------------- Phase 2: WMMA scores + quad softmax ----------------
    {
        const int pair = wave >> 1;              // (r,g) (g,b) (b,ir) (ir,g)
        const int gsel = wave & 1;               // heads gsel*4 .. gsel*4+3
        const int kvm[4] = {1, 2, 3, 1};
        const float* Q = &sQ[pair][0][0];
        const float* K = &sQ[kvm[pair]][0][0];

        const int m    = lane & 15;              // A row / B col index
        const int qk   = m & 3;                  // qi (rows) == ki (cols) on this lane
        const int hl   = m >> 2;                 // head within group
        const int koff = (lane >> 4) << 1;       // K-dim offset: 0 | 2
        const int cb   = (gsel * 4 + hl) * DH;

        v8f acc = {0.f, 0.f, 0.f, 0.f, 0.f, 0.f, 0.f, 0.f};
#pragma unroll
        for (int T = 0; T < 8; ++T) {            // 8 x K=4 chunks -> 32-dim contraction
            const int o = cb + T * 4 + koff;
            const float2 qa = *(const float2*)&Q[qk * CDIM + o];
            const float2 kb = *(const float2*)&K[qk * CDIM + o];
            v2f a;  a.x = qa.x;  a.y = qa.y;
            v2f bb; bb.x = kb.x; bb.y = kb.y;
            acc = __builtin_amdgcn_wmma_f32_16x16x4_f32(
                      false, a, false, bb, (short)0, acc, false, false);
        }

        const float SCALE = 0.17677669529663687f;   // 1/sqrt(32)
        const bool valid  = (lane < 16) ? (hl < 2) : (hl >= 2);
        const bool hsel   = (hl & 1) != 0;
        const int  hout   = gsel * 4 + hl;
        const int  ki     = lane & 3;

        const float a0 = acc[0], a1 = acc[1], a2 = acc[2], a3 = acc[3];
        const float a4 = acc[4], a5 = acc[5], a6 = acc[6], a7 = acc[7];
        const float s0 = (hsel ? a4 : a0) * SCALE;
        const float s1 = (hsel ? a5 : a1) * SCALE;
        const float s2 = (hsel ? a6 : a2) * SCALE;
        const float s3 = (hsel ? a7 : a3) * SCALE;
        const float p0 = quad_softmax(s0);
        const float p1 = quad_softmax(s1);
        const float p2 = quad_softmax(s2);
        const float p3 = quad_softmax(s3);
        if (valid) {
            sP[pair][hout][0][ki] = p0;
            sP[pair][hout][1][ki] = p1;
            sP[pair][hout][2][ki] = p2;
            sP[pair][hout][3][ki] = p3;
        }
    }
    __syncthreads();

    // ---------------- Phase 3: P@V + residual + LN + store (all b128) ----------------
    // Thread owns 4 consecutive channels of ONE pair: p = tid>>6, channels c4..c4+3.
    const int p  = tid >> 6;
    const int c4 = (tid & 63) << 2;
    const int h  = c4 >> 5;
    const int kvm[4] = {1, 2, 3, 1};

    float4 y4[NTOK];
    {
        const float* S = &sQ[p][0][0];
        const float* V = &sQ[kvm[p]][0][0];
        float4 vv[4], pm[4];
#pragma unroll
        for (int ki = 0; ki < 4; ++ki) vv[ki] = *(const float4*)&V[ki * CDIM + c4];
#pragma unroll
        for (int qi = 0; qi < 4; ++qi) pm[qi] = *(const float4*)&sP[p][h][qi][0];
#pragma unroll
        for (int qi = 0; qi < 4; ++qi) {
            float4 a = *(const float4*)&S[qi * CDIM + c4];
            a.x += pm[qi].x * vv[0].x + pm[qi].y * vv[1].x + pm[qi].z * vv[2].x + pm[qi].w * vv[3].x;
            a.y += pm[qi].x * vv[0].y + pm[qi].y * vv[1].y + pm[qi].z * vv[2].y + pm[qi].w * vv[3].y;
            a.z += pm[qi].x * vv[0].z + pm[qi].y * vv[1].z + pm[qi].z * vv[2].z + pm[qi].w * vv[3].z;
            a.w += pm[qi].x * vv[0].w + pm[qi].y * vv[1].w + pm[qi].z * vv[2].w + pm[qi].w * vv[3].w;
            y4[qi] = a;
        }
    }

    // LN reduction: per (p,qi) over 256 channels = 64 threads (waves 2p, 2p+1) x 4 chans.
#pragma unroll
    for (int qi = 0; qi < 4; ++qi) {
        const float4 a = y4[qi];
        float sy  = a.x + a.y + a.z + a.w;
        float sy2 = a.x * a.x + a.y * a.y + a.z * a.z + a.w * a.w;
        sy  = row_sum(sy);    // DPP-only, per 16-lane row
        sy2 = row_sum(sy2);
        if ((lane & 15) == 0) red2[wave][lane >> 4][qi] = make_float2(sy, sy2);
    }
    __syncthreads();

    const float* lwArr[4] = {lnw0, lnw1, lnw2, lnw3};
    const float* lbArr[4] = {lnb0, lnb1, lnb2, lnb3};
    const float4 lw4 = *(const float4*)(lwArr[p] + c4);
    const float4 lb4 = *(const float4*)(lbArr[p] + c4);

#pragma unroll
    for (int qi = 0; qi < 4; ++qi) {
        const float2 t0 = red2[2 * p + 0][0][qi];
        const float2 t1 = red2[2 * p + 0][1][qi];
        const float2 t2 = red2[2 * p + 1][0][qi];
        const float2 t3 = red2[2 * p + 1][1][qi];
        const float sy   = t0.x + t1.x + t2.x + t3.x;
        const float sy2  = t0.y + t1.y + t2.y + t3.y;
        const float mean = sy * (1.0f / 256.0f);
        const float var  = sy2 * (1.0f / 256.0f) - mean * mean;
        const float rstd = rsqrtf(var + 1e-5f);
        float4 o;
        o.x = (y4[qi].x - mean) * rstd * lw4.x + lb4.x;
        o.y = (y4[qi].y - mean) * rstd * lw4.y + lb4.y;
        o.z = (y4[qi].z - mean) * rstd * lw4.z + lb4.z;
        o.w = (y4[qi].w - mean) * rstd * lw4.w + lb4.w;
        *(float4*)(out + (long)tokBase[qi] * (4 * CDIM) + p * CDIM + c4) = o;
    }
}

extern "C" void kernel_launch(void* const* d_in, const int* in_sizes, int n_in,
                              void* d_out, int out_size, void* d_ws, size_t ws_size,
                              hipStream_t stream) {
    const float* r  = (const float*)d_in[0];
    const float* g  = (const float*)d_in[1];
    const float* b  = (const float*)d_in[2];
    const float* ir = (const float*)d_in[3];
    const float* ln1w = (const float*)d_in[4];
    const float* ln1b = (const float*)d_in[5];
    const float* ln2w = (const float*)d_in[6];
    const float* ln2b = (const float*)d_in[7];
    const float* ln3w = (const float*)d_in[8];
    const float* ln3b = (const float*)d_in[9];
    const float* ln4w = (const float*)d_in[10];
    const float* ln4b = (const float*)d_in[11];
    float* out = (float*)d_out;

    const int B = 8, H = 64, W = 64;                  // in_sizes[0] == B*H*W*256
    const int nWin = B * (H / WSIZE) * (W / WSIZE);   // 8192 windows
    swin_cross_attn_fused<<<dim3(nWin), dim3(256), 0, stream>>>(
        r, g, b, ir, ln1w, ln1b, ln2w, ln2b, ln3w, ln3b, ln4w, ln4b,
        out, B, H, W);
}